// LengthAdaptor_53206054863150
// MI455X (gfx1250) — compile-verified
//
#include <hip/hip_runtime.h>
#include <hip/hip_bf16.h>

// Problem constants (match reference)
#define B_      16
#define S_      256
#define D_      512
#define MAXDUR  10
#define TMAX    (S_ * MAXDUR)   // 2560
#define FRAMES  16               // output frames handled per gather block
#define GROUP   8                // frames per async batch (one ASYNCcnt drain per group)
#define NGROUPS (FRAMES / GROUP) // 2 -> each buffer set used exactly once per block
#define GTHREADS 128             // D_/4 float4 lanes per frame (4 waves)
#define QPF     (D_ / 4)         // float4s per frame = 128

// ---- CDNA5 async global<->LDS DMA availability probes -----------------------
#if __has_builtin(__builtin_amdgcn_global_load_async_to_lds_b128) && \
    __has_builtin(__builtin_amdgcn_global_store_async_from_lds_b128)
#define HAVE_ASYNC_LDS 1
#else
#define HAVE_ASYNC_LDS 0
#endif

// The builtins take pointers to a 16-byte int vector (per hipcc diagnostic).
typedef int v4i __attribute__((vector_size(4 * sizeof(int))));

__device__ __forceinline__ void wait_asynccnt0() {
#if __has_builtin(__builtin_amdgcn_s_wait_asynccnt)
    __builtin_amdgcn_s_wait_asynccnt(0);
#else
    asm volatile("s_wait_asynccnt 0x0" ::: "memory");
#endif
}

// -----------------------------------------------------------------------------
// Kernel 1: per-batch inclusive scan of durations (tiny: 16 x 256 ints).
// Writes cum[B,S] int32 into workspace, mel_len (as float) to output tail.
// -----------------------------------------------------------------------------
__global__ __launch_bounds__(S_) void la_scan_kernel(
    const int* __restrict__ dur, int* __restrict__ cum_ws,
    float* __restrict__ mel_out, int write_mel)
{
    __shared__ int s[S_];
    const int b   = blockIdx.x;
    const int tid = threadIdx.x;

    s[tid] = dur[b * S_ + tid];
    __syncthreads();
    // Hillis-Steele inclusive scan over 256 elements
    for (int off = 1; off < S_; off <<= 1) {
        int add = (tid >= off) ? s[tid - off] : 0;
        __syncthreads();
        s[tid] += add;
        __syncthreads();
    }
    cum_ws[b * S_ + tid] = s[tid];
    if (tid == S_ - 1 && write_mel)
        mel_out[b] = (float)s[tid];
}

// -----------------------------------------------------------------------------
// Kernel 2: gather/expand. Each block: one batch b, FRAMES contiguous output
// frames. Binary-search source token per frame, then stream the 2KB rows
// through LDS with the gfx1250 async DMA path: GROUP loads in flight per lane,
// one s_wait_asynccnt drain, then GROUP stores. Valid frames are a contiguous
// prefix (mel is block-uniform), so every async op runs with EXEC all-ones.
// Each group uses its own LDS buffer set, so there is no reuse hazard.
// -----------------------------------------------------------------------------
__global__ __launch_bounds__(GTHREADS) void la_gather_kernel(
    const float* __restrict__ x, const int* __restrict__ cum_ws,
    float* __restrict__ out)
{
    __shared__ int   s_cum[S_];
    __shared__ int   s_idx[FRAMES];
#if HAVE_ASYNC_LDS
    __shared__ float s_buf[NGROUPS][GROUP][D_];   // 2 x 8 x 2KB = 32KB staging
#endif

    const int tid = threadIdx.x;
    const int b   = blockIdx.y;
    const int t0  = blockIdx.x * FRAMES;

    for (int i = tid; i < S_; i += GTHREADS) s_cum[i] = cum_ws[b * S_ + i];
    __syncthreads();
    const int mel = s_cum[S_ - 1];

    // searchsorted(cum, t, side='right'): first i with cum[i] > t
    if (tid < FRAMES) {
        const int t = t0 + tid;
        int lo = 0, hi = S_;
        while (lo < hi) {
            int mid = (lo + hi) >> 1;
            if (s_cum[mid] <= t) lo = mid + 1; else hi = mid;
        }
        s_idx[tid] = (lo > S_ - 1) ? (S_ - 1) : lo;
    }
    __syncthreads();

    const float4* xb = (const float4*)(x + (size_t)b * S_ * D_);
    float4*       ob = (float4*)(out + (size_t)b * TMAX * D_);

#if HAVE_ASYNC_LDS
    for (int g = 0; g < NGROUPS; ++g) {
        // Phase 1: issue up to GROUP async global->LDS row copies (no waits).
        #pragma unroll
        for (int j = 0; j < GROUP; ++j) {
            const int f = g * GROUP + j;
            const int t = t0 + f;
            if (t < mel) {
                const float4* xrow = xb + (size_t)s_idx[f] * QPF;
                __builtin_amdgcn_global_load_async_to_lds_b128(
                    (v4i*)(xrow + tid), (v4i*)(&s_buf[g][j][0] + tid * 4), 0, 0);
            }
        }
        // Single per-wave drain: all this wave's group loads have landed in LDS.
        wait_asynccnt0();
        // Phase 2: issue async LDS->global stores (drained at final wait /
        // implicit S_WAIT_IDLE at S_ENDPGM). Invalid frames: plain zero fill.
        #pragma unroll
        for (int j = 0; j < GROUP; ++j) {
            const int f = g * GROUP + j;
            const int t = t0 + f;
            float4* orow = ob + (size_t)t * QPF;
            if (t < mel) {
                __builtin_amdgcn_global_store_async_from_lds_b128(
                    (v4i*)(orow + tid), (v4i*)(&s_buf[g][j][0] + tid * 4), 0, 0);
            } else {
                orow[tid] = make_float4(0.f, 0.f, 0.f, 0.f);
            }
        }
    }
    wait_asynccnt0();
#else
    for (int f = 0; f < FRAMES; ++f) {
        const int t = t0 + f;
        float4* orow = ob + (size_t)t * QPF;
        if (t < mel) {
            if (f + 1 < FRAMES && (t + 1) < mel)
                __builtin_prefetch(xb + (size_t)s_idx[f + 1] * QPF + tid, 0, 3);
            orow[tid] = (xb + (size_t)s_idx[f] * QPF)[tid];
        } else {
            orow[tid] = make_float4(0.f, 0.f, 0.f, 0.f);
        }
    }
#endif
}

// -----------------------------------------------------------------------------
// Launch: inputs are x [B,S,D] f32, duration [B,S] int. Output: out [B,TMAX,D]
// f32 followed by mel_len [B]. Workspace holds cum int32 [B,S] (16KB).
// -----------------------------------------------------------------------------
extern "C" void kernel_launch(void* const* d_in, const int* in_sizes, int n_in,
                              void* d_out, int out_size, void* d_ws, size_t ws_size,
                              hipStream_t stream) {
    const float* x   = (const float*)d_in[0];
    const int*   dur = (const int*)d_in[1];
    float*       out = (float*)d_out;
    int*      cum_ws = (int*)d_ws;

    const size_t out_main = (size_t)B_ * TMAX * D_;
    const int write_mel   = ((size_t)out_size > out_main) ? 1 : 0;
    float* mel_out        = out + out_main;

    la_scan_kernel<<<dim3(B_), dim3(S_), 0, stream>>>(dur, cum_ws, mel_out, write_mel);
    la_gather_kernel<<<dim3(TMAX / FRAMES, B_), dim3(GTHREADS), 0, stream>>>(x, cum_ws, out);
}